// forward_warp_max_motion_26869315404273
// MI455X (gfx1250) — compile-verified
//
#include <hip/hip_runtime.h>
#include <cstdint>

// Problem shape (fixed by setup_inputs): im0 [4,3,1080,1920] f32, flow [4,1080,1920,2] f32.
#define B_   4
#define C_   3
#define H_   1080
#define W_   1920
#define HW_  (H_ * W_)        // 2,073,600
#define N_   (B_ * HW_)       // 8,294,400   (divisible by 256)
#define OUT_ (B_ * C_ * HW_)  // 24,883,200  (divisible by 256)
#define EPS_ 1e-5f
#define DSCALE_ 1000.0f

// ---------------------------------------------------------------------------
// Pass 0: zero the depth buffer, the weight buffer and the output accumulator.
// (Harness poisons d_out / d_ws with 0xAA; we must re-init every call.)
// ---------------------------------------------------------------------------
__global__ __launch_bounds__(256) void fw_init(int* __restrict__ dbuf,
                                               float* __restrict__ wght,
                                               float* __restrict__ out) {
    int i = blockIdx.x * 256 + threadIdx.x;   // grid covers OUT_ exactly
    out[i] = 0.0f;
    if (i < N_) { dbuf[i] = 0; wght[i] = 0.0f; }
}

// ---------------------------------------------------------------------------
// Pass 1: scatter-max of quantized motion magnitude (the depth test).
// d >= 0 always, buffer init 0  ==  reference's  .at[idx].max(d) with init 0.
// ---------------------------------------------------------------------------
__global__ __launch_bounds__(256) void fw_depth(const float2* __restrict__ flow,
                                                int* __restrict__ dbuf) {
    int i  = blockIdx.x * 256 + threadIdx.x;  // [0, N_)
    int b  = i / HW_;
    int p  = i % HW_;
    int yy = p / W_;
    int xx = p % W_;

    float2 f = flow[i];
    float x = (float)xx + f.x;
    float y = (float)yy + f.y;
    int x0i = (int)floorf(x);
    int y0i = (int)floorf(y);
    int d   = __float2int_rn(sqrtf(f.x * f.x + f.y * f.y) * DSCALE_);

    int base = b * HW_;
#pragma unroll
    for (int cy = 0; cy < 2; ++cy) {
#pragma unroll
        for (int cx = 0; cx < 2; ++cx) {
            int tx = x0i + cx, ty = y0i + cy;
            if (tx >= 0 && tx < W_ && ty >= 0 && ty < H_)
                atomicMax(&dbuf[base + ty * W_ + tx], d);   // global_atomic_max_i32
        }
    }
}

// ---------------------------------------------------------------------------
// Pass 2: accumulate depth-test winners.  Flow is staged through LDS with the
// CDNA5 async-tensor data path (global_load_async_to_lds_b64 + s_wait_asynccnt)
// so each lane's VGPR pressure stays low while the atomics stream to L2.
// ---------------------------------------------------------------------------
__global__ __launch_bounds__(256) void fw_splat(const float*  __restrict__ im0,
                                                const float2* __restrict__ flow,
                                                const int*    __restrict__ dbuf,
                                                float* __restrict__ wght,
                                                float* __restrict__ out) {
    __shared__ float2 sflow[256];
    int tid = threadIdx.x;
    int i   = blockIdx.x * 256 + tid;         // [0, N_), exact grid => EXEC all-1s

    // Async copy this lane's float2 into its private LDS slot (8B per lane).
    // Low 32 bits of the flat shared address are the workgroup-relative LDS offset.
    unsigned      lds_off = (unsigned)(uintptr_t)(&sflow[tid]);
    const float2* gaddr   = flow + i;
    asm volatile("global_load_async_to_lds_b64 %0, %1, off"
                 :: "v"(lds_off), "v"(gaddr) : "memory");

    int b  = i / HW_;
    int p  = i % HW_;
    int yy = p / W_;
    int xx = p % W_;

    // Prefetch the three strided channel values while the async copy is in flight.
    const float* src = im0 + (size_t)b * (C_ * HW_) + p;
    __builtin_prefetch(src,             0, 0);   // global_prefetch_b8
    __builtin_prefetch(src + HW_,       0, 0);
    __builtin_prefetch(src + 2 * HW_,   0, 0);

    asm volatile("s_wait_asynccnt 0" ::: "memory");   // own-slot RAW: no barrier needed
    float2 f = sflow[tid];

    float x  = (float)xx + f.x;
    float y  = (float)yy + f.y;
    float x0 = floorf(x), y0 = floorf(y);
    float ax = x - x0,    ay = y - y0;
    int x0i  = (int)x0,   y0i = (int)y0;
    int d    = __float2int_rn(sqrtf(f.x * f.x + f.y * f.y) * DSCALE_);

    float c0 = src[0];
    float c1 = src[HW_];
    float c2 = src[2 * HW_];

    float wx[2] = {1.0f - ax, ax};
    float wy[2] = {1.0f - ay, ay};
    int   base  = b * HW_;
    float* obase = out + (size_t)b * (C_ * HW_);

#pragma unroll
    for (int cy = 0; cy < 2; ++cy) {
#pragma unroll
        for (int cx = 0; cx < 2; ++cx) {
            int tx = x0i + cx, ty = y0i + cy;
            if (tx < 0 || tx >= W_ || ty < 0 || ty >= H_) continue;
            int pt  = ty * W_ + tx;
            if (dbuf[base + pt] != d) continue;           // depth test (d >= 0 always)
            float w = wx[cx] * wy[cy];
            atomicAdd(&wght[base + pt], w);               // global_atomic_add_f32
            atomicAdd(&obase[pt],            w * c0);
            atomicAdd(&obase[pt + HW_],      w * c1);
            atomicAdd(&obase[pt + 2 * HW_],  w * c2);
        }
    }
}

// ---------------------------------------------------------------------------
// Pass 3: im1 = accum / max(wght, eps)   (clip(w, eps, None) == max(w, eps))
// ---------------------------------------------------------------------------
__global__ __launch_bounds__(256) void fw_norm(const float* __restrict__ wght,
                                               float* __restrict__ out) {
    int i = blockIdx.x * 256 + threadIdx.x;   // [0, OUT_)
    int b = i / (C_ * HW_);
    int p = i % HW_;                          // pixel within image plane
    float w = wght[b * HW_ + p];
    out[i] = out[i] / fmaxf(w, EPS_);
}

// ---------------------------------------------------------------------------
extern "C" void kernel_launch(void* const* d_in, const int* in_sizes, int n_in,
                              void* d_out, int out_size, void* d_ws, size_t ws_size,
                              hipStream_t stream) {
    const float*  im0  = (const float*)d_in[0];            // [4,3,1080,1920]
    const float2* flow = (const float2*)d_in[1];           // [4,1080,1920,2]
    float* out = (float*)d_out;                            // [4,3,1080,1920]

    // Workspace layout: dbuf (N_ int32) | wght (N_ f32)  => 66.4 MB total.
    int*   dbuf = (int*)d_ws;
    float* wght = (float*)((char*)d_ws + (size_t)N_ * sizeof(int));

    fw_init <<<OUT_ / 256, 256, 0, stream>>>(dbuf, wght, out);
    fw_depth<<<N_   / 256, 256, 0, stream>>>(flow, dbuf);
    fw_splat<<<N_   / 256, 256, 0, stream>>>(im0, flow, dbuf, wght, out);
    fw_norm <<<OUT_ / 256, 256, 0, stream>>>(wght, out);
}